// DCDicl_22376779612508
// MI455X (gfx1250) — compile-verified
//
#include <hip/hip_runtime.h>

typedef __attribute__((ext_vector_type(2))) float v2f;
typedef __attribute__((ext_vector_type(8))) float v8f;

#define NSAMP 4
#define CIN   64
#define COUT  4
#define DSZ   5
#define HH    96
#define WW    96
#define HW    (HH*WW)
#define NK    1600          // CIN * DSZ * DSZ
#define NB    32            // Cholesky block
#define CSTR  98            // LDS stride for corr tiles (even, odd/64*2 -> conflict-free)
#define SSTR  34            // LDS stride for syrk tiles

// ---------------------------------------------------------------------------
// corr[n][uv][a][b] = sum_{h,w} x[n,a,h+du,w+dv] * x[n,b,h,w]   (zero-padded)
// One workgroup per (uv, n): full 64x64 GEMM with K = 9216, staged row by row.
// ---------------------------------------------------------------------------
__global__ __launch_bounds__(256) void corr_kernel(const float* __restrict__ x,
                                                   float* __restrict__ corr) {
  const int uv = blockIdx.x;              // 0..80
  const int n  = blockIdx.y;
  const int du = uv / 9 - 4;
  const int dv = uv % 9 - 4;

  __shared__ float Asb[CIN * CSTR];
  __shared__ float Bsb[CIN * CSTR];

  const int tid  = threadIdx.x;
  const int wave = tid >> 5;
  const int lane = tid & 31;
  const int half = lane >> 4;             // 0/1: K pair select
  const int l16  = lane & 15;
  const int mt   = wave & 3;              // tile row 0..3
  const int nt   = (wave >> 2) * 2;       // tile col base (0 or 2)

  v8f acc0 = {0.f,0.f,0.f,0.f,0.f,0.f,0.f,0.f};
  v8f acc1 = {0.f,0.f,0.f,0.f,0.f,0.f,0.f,0.f};

  const float* xn = x + (size_t)n * CIN * HW;
  const int mrow  = mt * 16 + l16;
  const int nrow0 = (nt + 0) * 16 + l16;
  const int nrow1 = (nt + 1) * 16 + l16;

  for (int h = 0; h < HH; ++h) {
    const int  hs  = h + du;
    const bool hok = (hs >= 0) && (hs < HH);
    for (int t = tid; t < CIN * WW; t += 256) {
      const int a = t / WW;
      const int w = t - a * WW;
      Bsb[a * CSTR + w] = xn[(size_t)a * HW + h * WW + w];
      const int ws = w + dv;
      float av = 0.0f;
      if (hok && ws >= 0 && ws < WW) av = xn[(size_t)a * HW + hs * WW + ws];
      Asb[a * CSTR + w] = av;
    }
    __syncthreads();
#pragma unroll
    for (int k0 = 0; k0 < WW; k0 += 4) {
      const int kk = k0 + 2 * half;
      v2f a  = *(const v2f*)&Asb[mrow  * CSTR + kk];
      v2f b0 = *(const v2f*)&Bsb[nrow0 * CSTR + kk];
      v2f b1 = *(const v2f*)&Bsb[nrow1 * CSTR + kk];
      acc0 = __builtin_amdgcn_wmma_f32_16x16x4_f32(false, a, false, b0, (short)0, acc0, false, false);
      acc1 = __builtin_amdgcn_wmma_f32_16x16x4_f32(false, a, false, b1, (short)0, acc1, false, false);
    }
    __syncthreads();
  }

  float* cp = corr + ((size_t)n * 81 + uv) * CIN * CIN;
#pragma unroll
  for (int r = 0; r < 8; ++r) {
    const int row = mt * 16 + r + 8 * half;
    cp[(size_t)row * CIN + (nt + 0) * 16 + l16] = acc0[r];
    cp[(size_t)row * CIN + (nt + 1) * 16 + l16] = acc1[r];
  }
}

// ---------------------------------------------------------------------------
// Q[n, (cb,i,j), (ca,k,l)] = corr[n, (k-i+4)*9+(l-j+4), ca, cb] + diag reg
// ---------------------------------------------------------------------------
__global__ __launch_bounds__(256) void build_q_kernel(float* __restrict__ Q,
                                                      const float* __restrict__ corr,
                                                      const float* __restrict__ alpha,
                                                      const float* __restrict__ reg) {
  const size_t idx = (size_t)blockIdx.x * 256 + threadIdx.x;
  const size_t total = (size_t)NSAMP * NK * NK;
  if (idx >= total) return;
  const int n   = (int)(idx / ((size_t)NK * NK));
  const int rem = (int)(idx % ((size_t)NK * NK));
  const int kb  = rem / NK;
  const int ka  = rem % NK;
  const int cb = kb / 25, rb = kb % 25, ib = rb / 5, jb = rb % 5;
  const int ca = ka / 25, ra = ka % 25, ik = ra / 5, il = ra % 5;
  const int u = ik - ib + 4;
  const int v = il - jb + 4;
  float val = corr[(((size_t)n * 81 + u * 9 + v) * CIN + ca) * CIN + cb];
  if (kb == ka) {
    val += alpha[n] * (float)HW * reg[0] / (float)(DSZ * DSZ * CIN);
  }
  Q[idx] = val;
}

// ---------------------------------------------------------------------------
// P[n, (cin,i,j), co] = sum_{h,w} x[n,cin,h+i-2,w+j-2]*y[n,co,h,w] + a*d[...]
// One block per (ij, cin, n); 256-thread reduction over HW.
// ---------------------------------------------------------------------------
__global__ __launch_bounds__(256) void build_p_kernel(const float* __restrict__ x,
                                                      const float* __restrict__ y,
                                                      const float* __restrict__ dd,
                                                      const float* __restrict__ alpha,
                                                      const float* __restrict__ reg,
                                                      float* __restrict__ P) {
  const int ij  = blockIdx.x;   // 0..24
  const int cin = blockIdx.y;
  const int n   = blockIdx.z;
  const int di = ij / 5 - 2;
  const int dj = ij % 5 - 2;
  const float* xc = x + ((size_t)n * CIN + cin) * HW;
  const float* yn = y + (size_t)n * COUT * HW;
  float s0 = 0.f, s1 = 0.f, s2 = 0.f, s3 = 0.f;
  for (int t = threadIdx.x; t < HW; t += 256) {
    const int h = t / WW, w = t - h * WW;
    const int hs = h + di, ws = w + dj;
    if (hs < 0 || hs >= HH || ws < 0 || ws >= WW) continue;
    const float xv = xc[hs * WW + ws];
    s0 += xv * yn[0 * HW + t];
    s1 += xv * yn[1 * HW + t];
    s2 += xv * yn[2 * HW + t];
    s3 += xv * yn[3 * HW + t];
  }
  __shared__ float red[256][4];
  red[threadIdx.x][0] = s0; red[threadIdx.x][1] = s1;
  red[threadIdx.x][2] = s2; red[threadIdx.x][3] = s3;
  __syncthreads();
  for (int off = 128; off > 0; off >>= 1) {
    if (threadIdx.x < off) {
#pragma unroll
      for (int c = 0; c < 4; ++c) red[threadIdx.x][c] += red[threadIdx.x + off][c];
    }
    __syncthreads();
  }
  if (threadIdx.x < COUT) {
    const int co = threadIdx.x;
    const float a = alpha[n] * (float)HW * reg[0] / (float)(DSZ * DSZ * CIN);
    const int kb = cin * 25 + ij;
    P[((size_t)n * NK + kb) * COUT + co] =
        red[0][co] + a * dd[(((size_t)n * COUT + co) * CIN + cin) * 25 + ij];
  }
}

// ---------------------------------------------------------------------------
// Cholesky, blocked NB=32. (a) factor diagonal block, 1 wave per sample.
// ---------------------------------------------------------------------------
__global__ __launch_bounds__(32) void potrf_diag_kernel(float* __restrict__ Q, int j) {
  const int n = blockIdx.x;
  float* A = Q + (size_t)n * NK * NK;
  __shared__ float T[NB * 33];
  const int r = threadIdx.x;
  for (int c = 0; c < NB; ++c) T[r * 33 + c] = A[(size_t)(j + r) * NK + j + c];
  __syncthreads();
  for (int c = 0; c < NB; ++c) {
    const float s   = sqrtf(T[c * 33 + c]);
    const float inv = 1.0f / s;
    __syncthreads();
    if (r == c) T[c * 33 + c] = s;
    if (r > c)  T[r * 33 + c] *= inv;
    __syncthreads();
    if (r > c) {
      const float lrc = T[r * 33 + c];
      for (int c2 = c + 1; c2 <= r; ++c2) T[r * 33 + c2] -= lrc * T[c2 * 33 + c];
    }
    __syncthreads();
  }
  for (int c = 0; c <= r; ++c) A[(size_t)(j + r) * NK + j + c] = T[r * 33 + c];
}

// (b) panel solve: L21 = A21 * inv(L11^T); one row per thread.
__global__ __launch_bounds__(256) void trsm_kernel(float* __restrict__ Q, int j) {
  const int n = blockIdx.y;
  float* A = Q + (size_t)n * NK * NK;
  __shared__ float L11[NB * 33];
  for (int t = threadIdx.x; t < NB * NB; t += 256)
    L11[(t >> 5) * 33 + (t & 31)] = A[(size_t)(j + (t >> 5)) * NK + j + (t & 31)];
  __syncthreads();
  const int row = j + NB + blockIdx.x * 256 + threadIdx.x;
  if (row >= NK) return;
  float* ar = A + (size_t)row * NK + j;
  float v[NB];
#pragma unroll
  for (int c = 0; c < NB; ++c) {
    float s = ar[c];
#pragma unroll
    for (int m = 0; m < c; ++m) s -= v[m] * L11[c * 33 + m];
    v[c] = s / L11[c * 33 + c];
  }
#pragma unroll
  for (int c = 0; c < NB; ++c) ar[c] = v[c];
}

// (c) trailing update: A22 -= L21 * L21^T (lower block-triangle), WMMA f32.
__global__ __launch_bounds__(256) void syrk_kernel(float* __restrict__ Q, int j) {
  const int bm = blockIdx.x, bn = blockIdx.y;
  if (bm < bn) return;
  const int n = blockIdx.z;
  const int trail0 = j + NB;
  float* A = Q + (size_t)n * NK * NK;

  __shared__ float As[64 * SSTR];
  __shared__ float Bs[64 * SSTR];

  const int tid = threadIdx.x;
  const int rowm0 = trail0 + bm * 64;
  const int rown0 = trail0 + bn * 64;
  for (int t = tid; t < 64 * NB; t += 256) {
    const int rr = t >> 5;
    const int cc = t & 31;
    float av = 0.f, bv = 0.f;
    if (rowm0 + rr < NK) av = A[(size_t)(rowm0 + rr) * NK + j + cc];
    if (rown0 + rr < NK) bv = A[(size_t)(rown0 + rr) * NK + j + cc];
    As[rr * SSTR + cc] = av;
    Bs[rr * SSTR + cc] = bv;
  }
  __syncthreads();

  const int wave = tid >> 5, lane = tid & 31, half = lane >> 4, l16 = lane & 15;
  const int mt = wave & 3, nt = (wave >> 2) * 2;
  const int gc0 = rown0 + (nt + 0) * 16 + l16;
  const int gc1 = rown0 + (nt + 1) * 16 + l16;

  v8f acc0, acc1;
#pragma unroll
  for (int r = 0; r < 8; ++r) {
    const int grow = rowm0 + mt * 16 + r + 8 * half;
    acc0[r] = (grow < NK && gc0 < NK) ? A[(size_t)grow * NK + gc0] : 0.f;
    acc1[r] = (grow < NK && gc1 < NK) ? A[(size_t)grow * NK + gc1] : 0.f;
  }

  const int mrow  = mt * 16 + l16;
  const int nrow0 = (nt + 0) * 16 + l16;
  const int nrow1 = (nt + 1) * 16 + l16;
#pragma unroll
  for (int k0 = 0; k0 < NB; k0 += 4) {
    const int kk = k0 + 2 * half;
    v2f a  = *(const v2f*)&As[mrow  * SSTR + kk];
    a = -a;                                   // accumulate C + (-A)*B^T
    v2f b0 = *(const v2f*)&Bs[nrow0 * SSTR + kk];
    v2f b1 = *(const v2f*)&Bs[nrow1 * SSTR + kk];
    acc0 = __builtin_amdgcn_wmma_f32_16x16x4_f32(false, a, false, b0, (short)0, acc0, false, false);
    acc1 = __builtin_amdgcn_wmma_f32_16x16x4_f32(false, a, false, b1, (short)0, acc1, false, false);
  }

#pragma unroll
  for (int r = 0; r < 8; ++r) {
    const int grow = rowm0 + mt * 16 + r + 8 * half;
    if (grow < NK && gc0 < NK) A[(size_t)grow * NK + gc0] = acc0[r];
    if (grow < NK && gc1 < NK) A[(size_t)grow * NK + gc1] = acc1[r];
  }
}

// ---------------------------------------------------------------------------
// Forward solve L z = P (in place), blocked; one workgroup per sample.
// ---------------------------------------------------------------------------
__global__ __launch_bounds__(128) void solve_fwd_kernel(const float* __restrict__ Q,
                                                        float* __restrict__ P) {
  const int n = blockIdx.x;
  const float* A = Q + (size_t)n * NK * NK;
  float* Pn = P + (size_t)n * NK * COUT;
  __shared__ float Lb[NB * 33];
  __shared__ float zb[NB][COUT];
  const int tid = threadIdx.x;
  for (int jb = 0; jb < NK; jb += NB) {
    for (int t = tid; t < NB * NB; t += 128)
      Lb[(t >> 5) * 33 + (t & 31)] = A[(size_t)(jb + (t >> 5)) * NK + jb + (t & 31)];
    __syncthreads();
    if (tid < COUT) {
      const int co = tid;
      float zl[NB];
#pragma unroll
      for (int c = 0; c < NB; ++c) {
        float s = Pn[(size_t)(jb + c) * COUT + co];
#pragma unroll
        for (int m = 0; m < c; ++m) s -= Lb[c * 33 + m] * zl[m];
        zl[c] = s / Lb[c * 33 + c];
        zb[c][co] = zl[c];
        Pn[(size_t)(jb + c) * COUT + co] = zl[c];
      }
    }
    __syncthreads();
    for (int r = jb + NB + tid; r < NK; r += 128) {
      const float* Lr = A + (size_t)r * NK + jb;
      float a0 = 0.f, a1 = 0.f, a2 = 0.f, a3 = 0.f;
#pragma unroll
      for (int m = 0; m < NB; ++m) {
        const float lv = Lr[m];
        a0 += lv * zb[m][0]; a1 += lv * zb[m][1];
        a2 += lv * zb[m][2]; a3 += lv * zb[m][3];
      }
      float* pr = Pn + (size_t)r * COUT;
      pr[0] -= a0; pr[1] -= a1; pr[2] -= a2; pr[3] -= a3;
    }
    __syncthreads();
  }
}

// Backward solve L^T d = z (in place).
__global__ __launch_bounds__(128) void solve_bwd_kernel(const float* __restrict__ Q,
                                                        float* __restrict__ P) {
  const int n = blockIdx.x;
  const float* A = Q + (size_t)n * NK * NK;
  float* Pn = P + (size_t)n * NK * COUT;
  __shared__ float Lb[NB * 33];
  __shared__ float zb[NB][COUT];
  const int tid = threadIdx.x;
  for (int jb = NK - NB; jb >= 0; jb -= NB) {
    for (int t = tid; t < NB * NB; t += 128)
      Lb[(t >> 5) * 33 + (t & 31)] = A[(size_t)(jb + (t >> 5)) * NK + jb + (t & 31)];
    __syncthreads();
    if (tid < COUT) {
      const int co = tid;
      float dl[NB];
#pragma unroll
      for (int c = NB - 1; c >= 0; --c) {
        float s = Pn[(size_t)(jb + c) * COUT + co];
#pragma unroll
        for (int m = c + 1; m < NB; ++m) s -= Lb[m * 33 + c] * dl[m];
        dl[c] = s / Lb[c * 33 + c];
        zb[c][co] = dl[c];
        Pn[(size_t)(jb + c) * COUT + co] = dl[c];
      }
    }
    __syncthreads();
    for (int r = tid; r < jb; r += 128) {
      float a0 = 0.f, a1 = 0.f, a2 = 0.f, a3 = 0.f;
#pragma unroll
      for (int c = 0; c < NB; ++c) {
        const float lv = A[(size_t)(jb + c) * NK + r];   // L^T[r][jb+c]
        a0 += lv * zb[c][0]; a1 += lv * zb[c][1];
        a2 += lv * zb[c][2]; a3 += lv * zb[c][3];
      }
      float* pr = Pn + (size_t)r * COUT;
      pr[0] -= a0; pr[1] -= a1; pr[2] -= a2; pr[3] -= a3;
    }
    __syncthreads();
  }
}

// out[n][co][cin][i][j] = D[n][(cin,i,j)][co]
__global__ __launch_bounds__(256) void writeout_kernel(const float* __restrict__ P,
                                                       float* __restrict__ out) {
  const int o = blockIdx.x * 256 + threadIdx.x;
  const int total = NSAMP * COUT * CIN * 25;
  if (o >= total) return;
  const int n   = o / (COUT * CIN * 25);
  const int rem = o % (COUT * CIN * 25);
  const int co  = rem / (CIN * 25);
  const int r2  = rem % (CIN * 25);
  const int cin = r2 / 25;
  const int ij  = r2 % 25;
  out[o] = P[((size_t)n * NK + cin * 25 + ij) * COUT + co];
}

// ---------------------------------------------------------------------------
extern "C" void kernel_launch(void* const* d_in, const int* in_sizes, int n_in,
                              void* d_out, int out_size, void* d_ws, size_t ws_size,
                              hipStream_t stream) {
  const float* x     = (const float*)d_in[0];  // [4,1,64,96,96]
  const float* d     = (const float*)d_in[1];  // [4,4,64,5,5]
  const float* y     = (const float*)d_in[2];  // [4,4,1,96,96]
  const float* alpha = (const float*)d_in[3];  // [4]
  const float* reg   = (const float*)d_in[4];  // [1]
  float* out = (float*)d_out;

  float* Q    = (float*)d_ws;                         // N*K*K floats
  float* corr = Q + (size_t)NSAMP * NK * NK;          // N*81*64*64 floats
  float* P    = corr + (size_t)NSAMP * 81 * CIN * CIN; // N*K*COUT floats

  // 1) autocorrelation via fp32 WMMA
  corr_kernel<<<dim3(81, NSAMP), 256, 0, stream>>>(x, corr);

  // 2) assemble Q (+reg*I) and P (+reg*d)
  {
    const size_t total = (size_t)NSAMP * NK * NK;
    build_q_kernel<<<(unsigned)((total + 255) / 256), 256, 0, stream>>>(Q, corr, alpha, reg);
  }
  build_p_kernel<<<dim3(25, CIN, NSAMP), 256, 0, stream>>>(x, y, d, alpha, reg, P);

  // 3) blocked Cholesky (in place on Q, lower triangle)
  for (int j = 0; j < NK; j += NB) {
    potrf_diag_kernel<<<NSAMP, 32, 0, stream>>>(Q, j);
    const int rows = NK - j - NB;
    if (rows > 0) {
      trsm_kernel<<<dim3((rows + 255) / 256, NSAMP), 256, 0, stream>>>(Q, j);
      const int nt = (rows + 63) / 64;
      syrk_kernel<<<dim3(nt, nt, NSAMP), 256, 0, stream>>>(Q, j);
    }
  }

  // 4) triangular solves (in place on P) and output transpose
  solve_fwd_kernel<<<NSAMP, 128, 0, stream>>>(Q, P);
  solve_bwd_kernel<<<NSAMP, 128, 0, stream>>>(Q, P);
  writeout_kernel<<<(NSAMP * COUT * CIN * 25 + 255) / 256, 256, 0, stream>>>(P, out);
}